// CTRNNHebb_73100343378142
// MI455X (gfx1250) — compile-verified
//
#include <hip/hip_runtime.h>
#include <math.h>

typedef __attribute__((ext_vector_type(2))) float v2f;
typedef __attribute__((ext_vector_type(8))) float v8f;

#define T_STEPS 512
#define N_SAMP  256
#define HID     128
#define IN_DIM  64
#define H0_DIM  32
#define OUT_DIM 64

// ---------------------------------------------------------------------------
// Generic f32 WMMA GEMM: C[M,NC] = op(A[M,K]) @ B[K,NC] (+ bias), op = tanh?
// One 16x16 output tile per wave, 4 waves per block.
// A-matrix VGPR layout (V_WMMA_F32_16X16X4_F32): lane l holds row (l&15);
//   av.x = A[row, kk + 2*(l>>4)], av.y = A[row, kk + 2*(l>>4) + 1]
// B-matrix: bv.x = B[kk + 2*(l>>4), col], bv.y = B[kk + 2*(l>>4)+1, col]
// C/D: VGPR j, low-half lanes -> row j, high-half lanes -> row j+8.
// ---------------------------------------------------------------------------
template<int K, int NC, bool TANH_A, bool BIAS>
__global__ __launch_bounds__(128)
void gemm_wmma_f32(const float* __restrict__ A, const float* __restrict__ B,
                   const float* __restrict__ bias, float* __restrict__ C,
                   int Mtiles)
{
  const int lane  = threadIdx.x & 31;
  const int wid   = threadIdx.x >> 5;
  const int mtile = blockIdx.x * 4 + wid;
  const int ntile = blockIdx.y;
  if (mtile >= Mtiles) return;            // wave-uniform guard
  const int half = lane >> 4;
  const int l16  = lane & 15;
  const int rowA = mtile * 16 + l16;
  const int col  = ntile * 16 + l16;

  v8f c = {0.f,0.f,0.f,0.f,0.f,0.f,0.f,0.f};
  #pragma unroll
  for (int kk = 0; kk < K; kk += 4) {
    const float* ap = A + (size_t)rowA * K + kk + 2 * half;
    float a0 = ap[0];
    float a1 = ap[1];
    if (TANH_A) { a0 = tanhf(a0); a1 = tanhf(a1); }
    v2f av; av.x = a0; av.y = a1;
    v2f bv;
    bv.x = B[(size_t)(kk + 2 * half)     * NC + col];
    bv.y = B[(size_t)(kk + 2 * half + 1) * NC + col];
    // 8 args: (neg_a, A, neg_b, B, c_mod, C, reuse_a, reuse_b)
    c = __builtin_amdgcn_wmma_f32_16x16x4_f32(false, av, false, bv,
                                              (short)0, c, false, false);
  }
  const float bb = BIAS ? bias[col] : 0.f;
  #pragma unroll
  for (int j = 0; j < 8; ++j) {
    const int row = mtile * 16 + j + 8 * half;
    C[(size_t)row * NC + col] = c[j] + bb;
  }
}

// ---------------------------------------------------------------------------
// Recurrent Hebbian kernel: one workgroup (256 threads) per sample.
// Thread (k = tid&127, g = tid>>7) owns hebb[h, k] for h in [g*64, g*64+64),
// held in 64 VGPRs; W_rec / alpha_rec column slices also pinned in VGPRs.
// hebb never touches memory.
// ---------------------------------------------------------------------------
__global__ __launch_bounds__(256)
void rnn_hebb_kernel(const float* __restrict__ W_rec,
                     const float* __restrict__ alpha_rec,
                     const float* __restrict__ xin_all,  // (T,N,HID) = x@W_in + b_rec
                     float* __restrict__ h_all,          // (T,N,HID); slot 0 = h0
                     const float* __restrict__ alpha1,
                     const float* __restrict__ eta1)
{
  const int n     = blockIdx.x;
  const int tid   = threadIdx.x;
  const int k     = tid & (HID - 1);
  const int g     = tid >> 7;
  const int hbase = g * 64;

  __shared__ float r_s[HID];
  __shared__ float partial_s[2 * HID];
  __shared__ float hstate_s[HID];
  __shared__ float ern_s[HID];

  const float a   = alpha1[0];
  const float e   = eta1[0];
  const float oma = 1.f - a;
  const float ome = 1.f - e;

  float wr[64], ar[64], hb[64];
  #pragma unroll
  for (int i = 0; i < 64; ++i) {
    wr[i] = W_rec    [(size_t)(hbase + i) * HID + k];
    ar[i] = alpha_rec[(size_t)(hbase + i) * HID + k];
    hb[i] = 0.f;
  }
  if (tid < HID) hstate_s[tid] = h_all[(size_t)n * HID + tid];
  __syncthreads();

  for (int t = 1; t < T_STEPS; ++t) {
    if (tid < HID) r_s[tid] = tanhf(hstate_s[tid]);
    __syncthreads();                       // r_s visible

    // rec[k] = sum_h r[h] * (W_rec[h,k] + alpha_rec[h,k]*hebb[h,k])
    float acc0 = 0.f, acc1 = 0.f, acc2 = 0.f, acc3 = 0.f;
    #pragma unroll
    for (int i = 0; i < 64; i += 4) {
      acc0 = fmaf(r_s[hbase + i + 0], fmaf(ar[i + 0], hb[i + 0], wr[i + 0]), acc0);
      acc1 = fmaf(r_s[hbase + i + 1], fmaf(ar[i + 1], hb[i + 1], wr[i + 1]), acc1);
      acc2 = fmaf(r_s[hbase + i + 2], fmaf(ar[i + 2], hb[i + 2], wr[i + 2]), acc2);
      acc3 = fmaf(r_s[hbase + i + 3], fmaf(ar[i + 3], hb[i + 3], wr[i + 3]), acc3);
    }
    partial_s[tid] = (acc0 + acc1) + (acc2 + acc3);
    __syncthreads();                       // partials visible

    if (g == 0) {
      const float rec = partial_s[k] + partial_s[k + HID];
      const float xin = xin_all[((size_t)t * N_SAMP + n) * HID + k];
      const float hn  = oma * hstate_s[k] + a * (rec + xin);
      hstate_s[k] = hn;
      h_all[((size_t)t * N_SAMP + n) * HID + k] = hn;
      ern_s[k] = e * tanhf(hn);            // eta * r_new[k]
    }
    __syncthreads();                       // ern / hstate visible

    // hebb = (1-e)*hebb + r[h] * (e * r_new[k])
    const float ern = ern_s[k];
    #pragma unroll
    for (int i = 0; i < 64; ++i)
      hb[i] = fmaf(hb[i], ome, r_s[hbase + i] * ern);
    __syncthreads();                       // protect r_s WAR before next step
  }
}

// ---------------------------------------------------------------------------
// Inputs (setup_inputs order):
// 0 h0_data (1,N,32) | 1 input_ts (T,N,64) | 2 W_h0 (32,128) | 3 b_h0 (128)
// 4 W_in (64,128) | 5 W_rec (128,128) | 6 b_rec (128) | 7 alpha_rec (128,128)
// 8 W_out (128,64) | 9 alpha (1) | 10 eta (1)
// ---------------------------------------------------------------------------
extern "C" void kernel_launch(void* const* d_in, const int* in_sizes, int n_in,
                              void* d_out, int out_size, void* d_ws, size_t ws_size,
                              hipStream_t stream) {
  const float* h0_data   = (const float*)d_in[0];
  const float* input_ts  = (const float*)d_in[1];
  const float* W_h0      = (const float*)d_in[2];
  const float* b_h0      = (const float*)d_in[3];
  const float* W_in      = (const float*)d_in[4];
  const float* W_rec     = (const float*)d_in[5];
  const float* b_rec     = (const float*)d_in[6];
  const float* alpha_rec = (const float*)d_in[7];
  const float* W_out     = (const float*)d_in[8];
  const float* alpha     = (const float*)d_in[9];
  const float* eta       = (const float*)d_in[10];
  float* out = (float*)d_out;

  float* h_all   = (float*)d_ws;                              // T*N*HID f32 (64 MB)
  float* xin_all = h_all + (size_t)T_STEPS * N_SAMP * HID;    // T*N*HID f32 (64 MB)

  // 1) h0 = h0_data @ W_h0 + b_h0  -> h_all[t=0]   (256x32x128)
  gemm_wmma_f32<H0_DIM, HID, false, true>
      <<<dim3(N_SAMP / 64, HID / 16), 128, 0, stream>>>(
          h0_data, W_h0, b_h0, h_all, N_SAMP / 16);

  // 2) xin_all = input_ts @ W_in + b_rec  (131072x64x128), hoisted out of loop
  gemm_wmma_f32<IN_DIM, HID, false, true>
      <<<dim3((T_STEPS * N_SAMP) / 64, HID / 16), 128, 0, stream>>>(
          input_ts, W_in, b_rec, xin_all, (T_STEPS * N_SAMP) / 16);

  // 3) serial Hebbian recurrence, one workgroup per sample, hebb in VGPRs
  rnn_hebb_kernel<<<N_SAMP, 256, 0, stream>>>(
      W_rec, alpha_rec, xin_all, h_all, alpha, eta);

  // 4) out = tanh(h_all) @ W_out  (131072x128x64)
  gemm_wmma_f32<HID, OUT_DIM, true, false>
      <<<dim3((T_STEPS * N_SAMP) / 64, OUT_DIM / 16), 128, 0, stream>>>(
          h_all, W_out, nullptr, out, (T_STEPS * N_SAMP) / 16);
}